// iResNet_27745488732267
// MI455X (gfx1250) — compile-verified
//
#include <hip/hip_runtime.h>
#include <hip/hip_bf16.h>
#include <math.h>

// iResNet: x <- x + f(x) for L=4 blocks (3-layer leaky-ReLU MLP each), plus
// log|det J_f| at the updated x.  The SVD in the reference collapses
// analytically:  log|det J| = sum_k log|det w_k| + (#neg preacts) * log(0.1).
// So we need: 12 small LU log-dets (weights) + 5 GEMMs [4096,128]@[128,128]^T
// per layer, all fp32, done with V_WMMA_F32_16X16X4_F32.

#define B_ROWS 4096
#define D_DIM  128
#define L_LAYERS 4
#define SLOPE 0.1f
#define LDSTR 130   // padded LDS row stride (floats): even (float2-aligned), banks spread

typedef __attribute__((ext_vector_type(2))) float v2f;
typedef __attribute__((ext_vector_type(8))) float v8f;

// One wave computes a 16x16 f32 tile of  out = in(16x128) @ w(128x128)^T
// i.e. out[m,n] = sum_k in[m,k] * w[n,k].
// A-matrix (16x4): lane l<16 -> row l, k = kk+{0,1}; lane l>=16 -> row l-16, k = kk+{2,3}.
// B-matrix (4x16): B[k][n] = w[n][k]; same per-lane k pattern with n = lane&15.
__device__ __forceinline__ v8f wave_gemm16(const float* lds_in /*16 x LDSTR*/,
                                           const float* __restrict__ w /*[128][128]*/,
                                           int nb, int l16, int half)
{
    v8f acc = {};
    const float* arow = lds_in + l16 * LDSTR + 2 * half;
    const float* brow = w + (nb + l16) * D_DIM + 2 * half;
#pragma unroll
    for (int kk = 0; kk < D_DIM; kk += 4) {
        v2f a = *(const v2f*)(arow + kk);
        v2f b = *(const v2f*)(brow + kk);
        acc = __builtin_amdgcn_wmma_f32_16x16x4_f32(
            /*neg_a=*/false, a, /*neg_b=*/false, b,
            /*c_mod=*/(short)0, acc, /*reuse_a=*/false, /*reuse_b=*/false);
    }
    return acc;
}

// -------- LU log|det| of the 12 weight matrices (one block each) -----------
__global__ __launch_bounds__(128)
void lu_logdet_kernel(const float* __restrict__ W1, const float* __restrict__ W2,
                      const float* __restrict__ W3, float* __restrict__ wdet)
{
    __shared__ float A[D_DIM * D_DIM];   // 64 KB
    const int blk = blockIdx.x;          // 0..11
    const int l = blk / 3, j = blk % 3;
    const float* src = (j == 0 ? W1 : (j == 1 ? W2 : W3)) + (size_t)l * D_DIM * D_DIM;
    const int tid = threadIdx.x;

    for (int i = tid; i < D_DIM * D_DIM; i += 128) A[i] = src[i];
    __syncthreads();

    for (int k = 0; k < D_DIM - 1; ++k) {
        if (tid == 0) {
            // partial pivot (serial; tiny kernel) + row swap by thread 0
            int p = k; float best = fabsf(A[k * D_DIM + k]);
            for (int i = k + 1; i < D_DIM; ++i) {
                float v = fabsf(A[i * D_DIM + k]);
                if (v > best) { best = v; p = i; }
            }
            if (p != k) {
                for (int c = k; c < D_DIM; ++c) {
                    float t = A[k * D_DIM + c];
                    A[k * D_DIM + c] = A[p * D_DIM + c];
                    A[p * D_DIM + c] = t;
                }
            }
        }
        __syncthreads();
        if (tid > k) {
            float lf = A[tid * D_DIM + k] / A[k * D_DIM + k];
            for (int c = k + 1; c < D_DIM; ++c)
                A[tid * D_DIM + c] -= lf * A[k * D_DIM + c];
        }
        __syncthreads();
    }
    float term = logf(fabsf(A[tid * D_DIM + tid]));
    __syncthreads();
    A[tid] = term;           // reuse LDS as reduction scratch
    __syncthreads();
    if (tid == 0) {
        float s = 0.0f;
        for (int i = 0; i < D_DIM; ++i) s += A[i];
        wdet[blk] = s;
    }
}

// -------- One residual block: forward + analytic logdet terms --------------
__global__ __launch_bounds__(256)
void iresnet_layer_kernel(const float* __restrict__ xin, float* __restrict__ xout,
                          const float* __restrict__ W1, const float* __restrict__ B1,
                          const float* __restrict__ W2, const float* __restrict__ B2,
                          const float* __restrict__ W3, const float* __restrict__ B3,
                          const float* __restrict__ wdet,
                          float* __restrict__ logdet_out, int layer)
{
    __shared__ float bufX[16 * LDSTR];
    __shared__ float bufA[16 * LDSTR];
    __shared__ float bufB[16 * LDSTR];
    __shared__ int   negcnt[16];

    const int tid  = threadIdx.x;
    const int wave = tid >> 5;
    const int lane = tid & 31;
    const int half = lane >> 4;
    const int l16  = lane & 15;
    const int nb   = wave * 16;              // 8 waves cover N = 0..127
    const int mb   = blockIdx.x * 16;        // 16-row strip of the batch

    const float* w1 = W1 + (size_t)layer * D_DIM * D_DIM;
    const float* w2 = W2 + (size_t)layer * D_DIM * D_DIM;
    const float* w3 = W3 + (size_t)layer * D_DIM * D_DIM;
    const float* b1 = B1 + layer * D_DIM;
    const float* b2 = B2 + layer * D_DIM;
    const float* b3 = B3 + layer * D_DIM;

    // stage x strip into LDS
    for (int i = tid; i < 16 * D_DIM; i += 256) {
        int r = i >> 7, c = i & (D_DIM - 1);
        bufX[r * LDSTR + c] = xin[(size_t)(mb + r) * D_DIM + c];
    }
    if (tid < 16) negcnt[tid] = 0;
    __syncthreads();

    const int col = nb + l16;

    // S1: h1' = leaky(x @ w1^T + b1) -> bufA
    {
        v8f acc = wave_gemm16(bufX, w1, nb, l16, half);
        float bias = b1[col];
#pragma unroll
        for (int v = 0; v < 8; ++v) {
            float val = acc[v] + bias;
            bufA[(v + 8 * half) * LDSTR + col] = (val > 0.0f) ? val : SLOPE * val;
        }
    }
    __syncthreads();

    // S2: h2' = leaky(h1' @ w2^T + b2) -> bufB
    {
        v8f acc = wave_gemm16(bufA, w2, nb, l16, half);
        float bias = b2[col];
#pragma unroll
        for (int v = 0; v < 8; ++v) {
            float val = acc[v] + bias;
            bufB[(v + 8 * half) * LDSTR + col] = (val > 0.0f) ? val : SLOPE * val;
        }
    }
    __syncthreads();

    // S3: f = h2' @ w3^T + b3 ; x_new = x + f  -> global xout and bufA
    {
        v8f acc = wave_gemm16(bufB, w3, nb, l16, half);
        float bias = b3[col];
#pragma unroll
        for (int v = 0; v < 8; ++v) {
            int r = v + 8 * half;
            float xn = bufX[r * LDSTR + col] + (acc[v] + bias);
            xout[(size_t)(mb + r) * D_DIM + col] = xn;
            bufA[r * LDSTR + col] = xn;
        }
    }
    __syncthreads();

    // S4: g1 = x_new @ w1^T + b1 ; count nonpositive; leaky -> bufB
    {
        v8f acc = wave_gemm16(bufA, w1, nb, l16, half);
        float bias = b1[col];
#pragma unroll
        for (int v = 0; v < 8; ++v) {
            float val = acc[v] + bias;
            int r = v + 8 * half;
            if (val <= 0.0f) atomicAdd(&negcnt[r], 1);
            bufB[r * LDSTR + col] = (val > 0.0f) ? val : SLOPE * val;
        }
    }
    __syncthreads();

    // S5: g2 = leaky(g1) @ w2^T + b2 ; count nonpositive
    {
        v8f acc = wave_gemm16(bufB, w2, nb, l16, half);
        float bias = b2[col];
#pragma unroll
        for (int v = 0; v < 8; ++v) {
            float val = acc[v] + bias;
            if (val <= 0.0f) atomicAdd(&negcnt[v + 8 * half], 1);
        }
    }
    __syncthreads();

    // per-row logdet contribution for this layer
    if (tid < 16) {
        float wc = wdet[layer * 3 + 0] + wdet[layer * 3 + 1] + wdet[layer * 3 + 2];
        float term = (float)negcnt[tid] * logf(SLOPE) + wc;
        float prev = (layer == 0) ? 0.0f : logdet_out[mb + tid];
        logdet_out[mb + tid] = prev + term;
    }
}

extern "C" void kernel_launch(void* const* d_in, const int* in_sizes, int n_in,
                              void* d_out, int out_size, void* d_ws, size_t ws_size,
                              hipStream_t stream) {
    (void)in_sizes; (void)n_in; (void)out_size; (void)ws_size;
    const float* x  = (const float*)d_in[0];
    const float* W1 = (const float*)d_in[1];
    const float* B1 = (const float*)d_in[2];
    const float* W2 = (const float*)d_in[3];
    const float* B2 = (const float*)d_in[4];
    const float* W3 = (const float*)d_in[5];
    const float* B3 = (const float*)d_in[6];

    float* xout   = (float*)d_out;                       // [4096,128]
    float* logdet = xout + (size_t)B_ROWS * D_DIM;       // [4096]
    float* wdet   = (float*)d_ws;                        // 12 floats

    lu_logdet_kernel<<<12, 128, 0, stream>>>(W1, W2, W3, wdet);

    for (int layer = 0; layer < L_LAYERS; ++layer) {
        const float* xi = (layer == 0) ? x : xout;       // blocks touch only own rows
        iresnet_layer_kernel<<<B_ROWS / 16, 256, 0, stream>>>(
            xi, xout, W1, B1, W2, B2, W3, B3, wdet, logdet, layer);
    }
}